// ContrastMemory_13554916786346
// MI455X (gfx1250) — compile-verified
//
#include <hip/hip_runtime.h>

typedef __attribute__((ext_vector_type(2))) float v2f;
typedef __attribute__((ext_vector_type(4))) float v4f;
typedef __attribute__((ext_vector_type(8))) float v8f;

#define BATCH 128
#define DIM 512
#define KP 2048            // K + NPOS
#define NROWS1 16084
#define NROWS2 604731
#define T_INV (1.0f / 0.07f)

// ---------------------------------------------------------------------------
// 1) Bulk copy of both memory banks into the output (b128 grid-stride).
//    m1 (33 MB) copied with regular temporal policy -> stays in 192MB L2 for
//    the gather kernel. m2 (1.24 GB) is streaming -> non-temporal both ways.
// ---------------------------------------------------------------------------
__global__ void copy_mem_kernel(const float* __restrict__ m1, const float* __restrict__ m2,
                                float* __restrict__ o1, float* __restrict__ o2) {
  const long n1 = (long)NROWS1 * DIM / 4;
  const long n2 = (long)NROWS2 * DIM / 4;
  long i = (long)blockIdx.x * blockDim.x + threadIdx.x;
  long stride = (long)gridDim.x * blockDim.x;
  const v4f* s1 = (const v4f*)m1; v4f* d1 = (v4f*)o1;
  const v4f* s2 = (const v4f*)m2; v4f* d2 = (v4f*)o2;
  for (long j = i; j < n1; j += stride) d1[j] = s1[j];
  for (long j = i; j < n2; j += stride) {
    v4f t = __builtin_nontemporal_load(&s2[j]);
    __builtin_nontemporal_store(t, &d2[j]);
  }
}

// ---------------------------------------------------------------------------
// 2) Momentum update of the B touched rows (reads OLD memory, writes output).
//    Last-wins semantics for duplicate row indices.
// ---------------------------------------------------------------------------
__global__ void __launch_bounds__(256) momentum_update_kernel(
    const float* __restrict__ mem, const int* __restrict__ y,
    const float* __restrict__ v, float* __restrict__ out) {
  int b = blockIdx.x;
  int yb = y[b];
  for (int b2 = b + 1; b2 < BATCH; ++b2)
    if (y[b2] == yb) return;           // a later batch overwrites this row
  __shared__ float red[256];
  int t = threadIdx.x;
  long rbase = (long)yb * DIM;
  float p0 = 0.5f * mem[rbase + t]       + 0.5f * v[b * DIM + t];
  float p1 = 0.5f * mem[rbase + t + 256] + 0.5f * v[b * DIM + t + 256];
  red[t] = p0 * p0 + p1 * p1;
  __syncthreads();
  for (int o = 128; o > 0; o >>= 1) {
    if (t < o) red[t] += red[t + o];
    __syncthreads();
  }
  float invn = 1.0f / sqrtf(red[0]);
  out[rbase + t]       = p0 * invn;
  out[rbase + t + 256] = p1 * invn;
}

// ---------------------------------------------------------------------------
// 3) Contrast logits via V_WMMA_F32_16X16X4_F32 + in-block logsumexp.
//    One block (8 wave32) per batch b. B fragments (v1/v2/zero rows) are
//    staged in LDS so the inner loop has NO exec-mask divergence:
//      global_load_b64 (A) + ds_load_b64 (B) + v_wmma, two independent
//    accumulators per wave iteration to break the WMMA dependency chain.
//    A layout: lane m=lane&15 holds row M=m, K-pair (lane>=16 ? {2,3}:{0,1}).
//    B layout: column N=lane&15 (N=0 -> v1 row, N=1 -> v2 row, else zero).
//    D layout: VGPR i = M=i (lanes 0-15) / M=i+8 (lanes 16-31), N=lane&15.
// ---------------------------------------------------------------------------
__global__ void __launch_bounds__(256) contrast_logits_kernel(
    const float* __restrict__ v1, const float* __restrict__ v2,
    const int* __restrict__ idx1, const int* __restrict__ idx2,
    const float* __restrict__ m1, const float* __restrict__ m2,
    float* __restrict__ partial) {
  __shared__ float slog[2 * 2 * KP];   // logits[t][c], 8192 floats = 32 KB
  __shared__ float vlds[3 * DIM];      // v1 row | v2 row | zero row (6 KB)
  __shared__ float red[256];

  const int b    = blockIdx.x;
  const int tid  = threadIdx.x;
  const int wave = tid >> 5;
  const int lane = tid & 31;
  const int n      = lane & 15;        // N column == A row within tile
  const int khalf  = lane >> 4;        // 0: K={0,1}, 1: K={2,3}
  const int coloff = khalf * 2;

  // stage B source rows into LDS (zero row for unused N columns)
  for (int d = tid; d < DIM; d += 256) {
    vlds[d]           = v1[b * DIM + d];
    vlds[DIM + d]     = v2[b * DIM + d];
    vlds[2 * DIM + d] = 0.0f;
  }
  __syncthreads();

  const int sel = (n < 2) ? n : 2;
  const float* brow = vlds + sel * DIM + coloff;   // unconditional per-lane B

  // wave w owns tiles {w, w+8, ..., w+248}; process them in pairs (t, t+8)
  for (int j = 0; j < 16; ++j) {
    const int tile0 = wave + 16 * j;
    const int tile1 = tile0 + 8;
    const int c0 = tile0 * 16, c1 = tile1 * 16;
    const int candA = c0 + n,  candB = c1 + n;

    const float *memA, *memB;
    int rowA, rowB;
    if (candA < KP) { rowA = idx1[b * KP + candA];      memA = m1; }
    else            { rowA = idx2[b * KP + candA - KP]; memA = m2; }
    if (candB < KP) { rowB = idx1[b * KP + candB];      memB = m1; }
    else            { rowB = idx2[b * KP + candB - KP]; memB = m2; }
    const float* arowA = memA + (long)rowA * DIM + coloff;
    const float* arowB = memB + (long)rowB * DIM + coloff;

    v8f accA = {0.f, 0.f, 0.f, 0.f, 0.f, 0.f, 0.f, 0.f};
    v8f accB = {0.f, 0.f, 0.f, 0.f, 0.f, 0.f, 0.f, 0.f};
#pragma unroll 4
    for (int k = 0; k < DIM; k += 4) {
      v2f aA = *(const v2f*)(arowA + k);
      v2f aB = *(const v2f*)(arowB + k);
      v2f bb = *(const v2f*)(brow + k);            // ds_load_b64, no divergence
      accA = __builtin_amdgcn_wmma_f32_16x16x4_f32(
          false, aA, false, bb, (short)0, accA, false, false);
      accB = __builtin_amdgcn_wmma_f32_16x16x4_f32(
          false, aB, false, bb, (short)0, accB, false, false);
    }
    if (n < 2) {                       // lanes 0,1,16,17 hold valid columns
      const int t = n;
      const int mbase = khalf * 8;
#pragma unroll
      for (int i = 0; i < 8; ++i) {
        slog[t * (2 * KP) + c0 + mbase + i] = accA[i] * T_INV;
        slog[t * (2 * KP) + c1 + mbase + i] = accB[i] * T_INV;
      }
    }
  }
  __syncthreads();

  // per-row (t,b) logsumexp over 4096 logits, then positive-column log-probs
  float accb = 0.0f;                   // meaningful on tid==0 only
  for (int t = 0; t < 2; ++t) {
    const float* rowp = slog + t * (2 * KP);
    float mx = -__builtin_inff();
    for (int c = tid; c < 2 * KP; c += 256) mx = fmaxf(mx, rowp[c]);
    red[tid] = mx;
    __syncthreads();
    for (int o = 128; o > 0; o >>= 1) {
      if (tid < o) red[tid] = fmaxf(red[tid], red[tid + o]);
      __syncthreads();
    }
    mx = red[0];
    __syncthreads();
    float s = 0.0f;
    for (int c = tid; c < 2 * KP; c += 256) s += expf(rowp[c] - mx);
    red[tid] = s;
    __syncthreads();
    for (int o = 128; o > 0; o >>= 1) {
      if (tid < o) red[tid] += red[tid + o];
      __syncthreads();
    }
    if (tid == 0) {
      float lse = mx + logf(red[0]);
      accb += 0.5f * ((rowp[0] - lse) + (rowp[KP] - lse));
    }
    __syncthreads();
  }
  if (tid == 0) partial[b] = accb;
}

// ---------------------------------------------------------------------------
// 4) Deterministic fixed-order final reduction of the 128 batch partials.
// ---------------------------------------------------------------------------
__global__ void final_loss_kernel(const float* __restrict__ partial,
                                  float* __restrict__ loss) {
  if (threadIdx.x == 0 && blockIdx.x == 0) {
    float s = 0.0f;
    for (int b = 0; b < BATCH; ++b) s += partial[b];
    loss[0] = -s / (2.0f * BATCH);     // T/BASE_T == 1
  }
}

extern "C" void kernel_launch(void* const* d_in, const int* in_sizes, int n_in,
                              void* d_out, int out_size, void* d_ws, size_t ws_size,
                              hipStream_t stream) {
  const float* v1   = (const float*)d_in[0];
  const int*   y1   = (const int*)d_in[1];
  const float* v2   = (const float*)d_in[2];
  const int*   y2   = (const int*)d_in[3];
  const int*   idx1 = (const int*)d_in[4];
  const int*   idx2 = (const int*)d_in[5];
  const float* m1   = (const float*)d_in[6];
  const float* m2   = (const float*)d_in[7];

  float* out  = (float*)d_out;
  float* loss = out;
  float* o1   = out + 1;
  float* o2   = o1 + (long)NROWS1 * DIM;
  float* partial = (float*)d_ws;       // 128 floats of scratch

  copy_mem_kernel<<<16384, 256, 0, stream>>>(m1, m2, o1, o2);
  momentum_update_kernel<<<BATCH, 256, 0, stream>>>(m1, y1, v1, o1);
  momentum_update_kernel<<<BATCH, 256, 0, stream>>>(m2, y2, v2, o2);
  contrast_logits_kernel<<<BATCH, 256, 0, stream>>>(v1, v2, idx1, idx2, m1, m2, partial);
  final_loss_kernel<<<1, 64, 0, stream>>>(partial, loss);
}